// Plenoxels_40681930227959
// MI455X (gfx1250) — compile-verified
//
#include <hip/hip_runtime.h>
#include <hip/hip_bf16.h>
#include <stdint.h>

#define NLV 192
#define BLOCK 64            // 2 waves of 32
#define SLOT_F4 57          // 8 corners * 7 float4 = 56, +1 float4 pad -> 912 B/lane (bank-conflict-free)

// SH constants
#define SH_C0   0.28209479177387814f
#define SH_C1   0.4886025119029199f
#define SH_C2_0 1.0925484305920792f
#define SH_C2_1 (-1.0925484305920792f)
#define SH_C2_2 0.31539156525252005f
#define SH_C2_3 (-1.0925484305920792f)
#define SH_C2_4 0.5462742152960396f

// Async global->LDS: IOFFSET is applied to BOTH the LDS dest and the global source
// (CDNA5 ISA), so 7 chunks per corner share one LDS base VGPR and
// one global address pair.
#define ALOAD(LA, GA, OFF) \
    asm volatile("global_load_async_to_lds_b128 %0, %1, off offset:" OFF \
                 :: "v"(LA), "v"(GA) : "memory")

// Async loads return "done" in order -> incremental waits pipeline FMAs against
// still-in-flight gathers.
#define AWAIT(NSTR) asm volatile("s_wait_asynccnt " NSTR ::: "memory")

__global__ __launch_bounds__(BLOCK)
__attribute__((amdgpu_waves_per_eu(4)))
void plenoxels_kernel(const float* __restrict__ xin,
                      const float* __restrict__ din,
                      const float* __restrict__ voxel,
                      float* __restrict__ color,
                      float* __restrict__ sigma,
                      int N)
{
    // Per-lane contiguous slot: 57 float4 (912 B). 64 lanes -> 58,368 B.
    __shared__ float4 lds_buf[BLOCK * SLOT_F4];

    const int tid = blockIdx.x * BLOCK + threadIdx.x;
    const int i   = (tid < N) ? tid : (N - 1);   // clamp: EXEC stays all-ones

    // ---- position -> corner indices + weights ----
    const float x0 = xin[3 * i + 0];
    const float x1 = xin[3 * i + 1];
    const float x2 = xin[3 * i + 2];

    const float s0 = x0 * (1.0f / 1.5f);
    const float s1 = x1 * (1.0f / 1.5f);
    const float s2 = x2 * (1.0f / 1.5f);

    const bool mask = (fabsf(s0) < 0.5f) && (fabsf(s1) < 0.5f) && (fabsf(s2) < 0.5f);

    const float id0 = fminf(fmaxf(s0 * 96.0f + 96.0f, 0.0f), 191.0f);
    const float id1 = fminf(fmaxf(s1 * 96.0f + 96.0f, 0.0f), 191.0f);
    const float id2 = fminf(fmaxf(s2 * 96.0f + 96.0f, 0.0f), 191.0f);

    const float f0f = floorf(id0), f1f = floorf(id1), f2f = floorf(id2);
    const int f0 = (int)f0f,        f1 = (int)f1f,        f2 = (int)f2f;
    const int c0 = (int)ceilf(id0), c1 = (int)ceilf(id1), c2 = (int)ceilf(id2);

    const float t0 = id0 - f0f, t1 = id1 - f1f, t2 = id2 - f2f;
    const float u0 = 1.0f - t0, u1 = 1.0f - t1, u2 = 1.0f - t2;

    const int bxf = f0 * (NLV * NLV * 28), bxc = c0 * (NLV * NLV * 28);
    const int byf = f1 * (NLV * 28),       byc = c1 * (NLV * 28);
    const int bzf = f2 * 28,               bzc = c2 * 28;

    // Corner order exactly as the reference stacks (vx, vy, vz):
    int base[8];
    base[0] = bxf + byf + bzf;   // (f,f,f)
    base[1] = bxc + byf + bzf;   // (c,f,f)
    base[2] = bxf + byc + bzf;   // (f,c,f)
    base[3] = bxf + byf + bzc;   // (f,f,c)
    base[4] = bxf + byc + bzc;   // (f,c,c)
    base[5] = bxc + byf + bzc;   // (c,f,c)
    base[6] = bxc + byc + bzf;   // (c,c,f)
    base[7] = bxc + byc + bzc;   // (c,c,c)

    float w[8];
    w[0] = u2 * u1 * u0;
    w[1] = u2 * u1 * t0;
    w[2] = u2 * t1 * u0;
    w[3] = u2 * t1 * t0;
    w[4] = t2 * u1 * u0;
    w[5] = t2 * u1 * t0;
    w[6] = t2 * t1 * u0;
    w[7] = t2 * t1 * t0;

    // ---- issue all 56 async gathers first (max MLP, zero VGPR landing slots) ----
    // Generic LDS pointer: low 32 bits are the LDS byte offset.
    const unsigned lds_lane0 =
        (unsigned)(size_t)(&lds_buf[0]) + (unsigned)(threadIdx.x * (SLOT_F4 * 16));

#pragma unroll
    for (int j = 0; j < 8; ++j) {
        const unsigned la = lds_lane0 + (unsigned)(j * 112);   // 7 float4 per corner
        const unsigned long long ga = (unsigned long long)(size_t)(voxel + base[j]);
        ALOAD(la, ga, "0");
        ALOAD(la, ga, "16");
        ALOAD(la, ga, "32");
        ALOAD(la, ga, "48");
        ALOAD(la, ga, "64");
        ALOAD(la, ga, "80");
        ALOAD(la, ga, "96");
    }

    // ---- pipelined accumulation: corner j ready once asynccnt <= 49 - 7j ----
    float acc[28];
#pragma unroll
    for (int c = 0; c < 28; ++c) acc[c] = 0.0f;

    const int rb = threadIdx.x * SLOT_F4;

#define CORNER(J, WAITN)                                        \
    do {                                                        \
        AWAIT(WAITN);                                           \
        const float wj = w[J];                                  \
        _Pragma("unroll")                                       \
        for (int q = 0; q < 7; ++q) {                           \
            const float4 v = lds_buf[rb + (J) * 7 + q];         \
            acc[4 * q + 0] = fmaf(wj, v.x, acc[4 * q + 0]);     \
            acc[4 * q + 1] = fmaf(wj, v.y, acc[4 * q + 1]);     \
            acc[4 * q + 2] = fmaf(wj, v.z, acc[4 * q + 2]);     \
            acc[4 * q + 3] = fmaf(wj, v.w, acc[4 * q + 3]);     \
        }                                                       \
    } while (0)

    CORNER(0, "49");
    CORNER(1, "42");
    CORNER(2, "35");
    CORNER(3, "28");
    CORNER(4, "21");
    CORNER(5, "14");
    CORNER(6, "7");
    CORNER(7, "0");
#undef CORNER

    // ---- sigma + SH color ----
    const float dx = din[3 * i + 0];
    const float dy = din[3 * i + 1];
    const float dz = din[3 * i + 2];
    const float xx = dx * dx, yy = dy * dy, zz = dz * dz;

    const float sg = mask ? fmaxf(acc[0], 0.0f) : 0.0f;

    float col[3];
#pragma unroll
    for (int ch = 0; ch < 3; ++ch) {
        const float* k = &acc[1 + ch * 9];
        float r = SH_C0 * k[0]
                - SH_C1 * dy * k[1]
                + SH_C1 * dz * k[2]
                - SH_C1 * dx * k[3]
                + SH_C2_0 * (dx * dy) * k[4]
                + SH_C2_1 * (dy * dz) * k[5]
                + SH_C2_2 * (2.0f * zz - xx - yy) * k[6]
                + SH_C2_3 * (dx * dz) * k[7]
                + SH_C2_4 * (xx - yy) * k[8];
        col[ch] = mask ? r : 0.0f;
    }

    if (tid < N) {
        color[3 * tid + 0] = col[0];
        color[3 * tid + 1] = col[1];
        color[3 * tid + 2] = col[2];
        sigma[tid] = sg;
    }
}

extern "C" void kernel_launch(void* const* d_in, const int* in_sizes, int n_in,
                              void* d_out, int out_size, void* d_ws, size_t ws_size,
                              hipStream_t stream) {
    const float* x     = (const float*)d_in[0];
    const float* d     = (const float*)d_in[1];
    const float* voxel = (const float*)d_in[2];

    const int N = in_sizes[0] / 3;

    float* out   = (float*)d_out;
    float* color = out;                     // (N,3) flat
    float* sigma = out + (size_t)3 * N;     // (N,)

    dim3 block(BLOCK);
    dim3 grid((N + BLOCK - 1) / BLOCK);
    plenoxels_kernel<<<grid, block, 0, stream>>>(x, d, voxel, color, sigma, N);
}